// GATLayer_82772609728558
// MI455X (gfx1250) — compile-verified
//
#include <hip/hip_runtime.h>
#include <hip/hip_bf16.h>
#include <math.h>

#define D_DIM 128
#define NEG_SLOPE 0.2f

typedef __attribute__((ext_vector_type(2))) float v2f;
typedef __attribute__((ext_vector_type(8))) float v8f;

// ---------------------------------------------------------------------------
// Kernel 1: init out=0, denom=0, m=-inf
// ---------------------------------------------------------------------------
__global__ void gat_init_kernel(float* __restrict__ out, int out_n,
                                float* __restrict__ m, float* __restrict__ denom,
                                int n_nodes) {
    int idx = blockIdx.x * blockDim.x + threadIdx.x;
    if (idx < out_n) out[idx] = 0.0f;
    if (idx < n_nodes) {
        m[idx] = -__builtin_inff();
        denom[idx] = 0.0f;
    }
}

// ---------------------------------------------------------------------------
// Kernel 2: s1 = h @ a1, s2 = h @ a2 via V_WMMA_F32_16X16X4_F32.
// One wave handles 16 nodes (rows of h). A = 16x4 tile of h, B = 4x16 with
// column 0 = a1 slice, column 1 = a2 slice, rest zero. K-loop: 128/4 = 32 WMMAs.
// Branch-free inner loop: masked float2 (b64) loads, no EXEC churn around WMMA.
//
// A layout (32-bit A 16x4): lanes 0-15 -> M=lane, K={0,1}; lanes 16-31 -> K={2,3}.
// B layout (32-bit B 4x16): lanes 0-15 -> N=lane, K={0,1}; lanes 16-31 -> K={2,3}.
// C/D layout (32-bit 16x16): VGPR v: lanes 0-15 -> M=v, lanes 16-31 -> M=v+8.
// ---------------------------------------------------------------------------
__global__ void gat_gemv_wmma_kernel(const float* __restrict__ h,
                                     const float* __restrict__ a_w,
                                     float* __restrict__ s1,
                                     float* __restrict__ s2,
                                     int n_nodes) {
    int wave = (blockIdx.x * blockDim.x + threadIdx.x) >> 5;
    int lane = threadIdx.x & 31;
    int base = wave * 16;
    if (base >= n_nodes) return;           // wave-uniform: EXEC stays all-ones

    int mrow  = lane & 15;                 // A row within tile
    int khalf = (lane >> 4) * 2;           // 0 for lanes 0-15, 2 for lanes 16-31
    int ncol  = lane & 15;                 // B/N column this lane owns

    // A operand: clamped row pointer + 0/1 mask (no divergent branch)
    int   row    = base + mrow;
    int   row_c  = (row < n_nodes) ? row : (n_nodes - 1);
    float amask  = (row < n_nodes) ? 1.0f : 0.0f;
    const float2* ap = (const float2*)(h + (size_t)row_c * D_DIM + khalf);

    // B operand: lanes ncol>=2 read the ncol==0 column (valid addr), masked to 0
    float bmask  = (ncol < 2) ? 1.0f : 0.0f;
    const float2* bp = (const float2*)(a_w + ((ncol == 1) ? D_DIM : 0) + khalf);

    v8f acc = {};
    #pragma unroll 4
    for (int k = 0; k < D_DIM; k += 4) {
        float2 av = ap[k >> 1];            // global_load_b64
        float2 bv = bp[k >> 1];            // global_load_b64 (broadcast-ish, L0 hit)
        v2f a, b;
        a.x = av.x * amask;  a.y = av.y * amask;
        b.x = bv.x * bmask;  b.y = bv.y * bmask;
        acc = __builtin_amdgcn_wmma_f32_16x16x4_f32(
            /*neg_a=*/false, a, /*neg_b=*/false, b,
            /*c_mod=*/(short)0, acc, /*reuse_a=*/false, /*reuse_b=*/false);
    }

    // Lanes with N==0 hold s1 for 8 nodes, N==1 hold s2 for 8 nodes.
    int mbase = (lane >> 4) * 8;           // lanes 0-15: M=v, lanes 16-31: M=v+8
    if (ncol < 2) {
        float* sout = (ncol == 0) ? s1 : s2;
        #pragma unroll
        for (int v = 0; v < 8; ++v) {
            int node = base + mbase + v;
            if (node < n_nodes) sout[node] = acc[v];
        }
    }
}

// ---------------------------------------------------------------------------
// Float atomic max with the sign-split int/uint trick (monotone encodings).
// Init value -inf (0xFF800000) is correct for both paths.
// ---------------------------------------------------------------------------
__device__ __forceinline__ void atomic_max_float(float* addr, float val) {
    if (val >= 0.0f) {
        atomicMax((int*)addr, __float_as_int(val));
    } else {
        atomicMin((unsigned int*)addr, __float_as_uint(val));
    }
}

// ---------------------------------------------------------------------------
// Kernel 3: e = leaky_relu(s1[src] + s2[dst]); segment max into m[dst]
// ---------------------------------------------------------------------------
__global__ void gat_edge_e_kernel(const float* __restrict__ s1,
                                  const float* __restrict__ s2,
                                  const int* __restrict__ src,
                                  const int* __restrict__ dst,
                                  float* __restrict__ e,
                                  float* __restrict__ m,
                                  int n_edges) {
    int i = blockIdx.x * blockDim.x + threadIdx.x;
    if (i >= n_edges) return;
    float x = s1[src[i]] + s2[dst[i]];
    float ev = (x > 0.0f) ? x : NEG_SLOPE * x;
    e[i] = ev;
    atomic_max_float(&m[dst[i]], ev);
}

// ---------------------------------------------------------------------------
// Kernel 4: ex = exp(e - m[dst]) (in place over e); denom[dst] += ex.
// m[dst] is finite for every edge's dst by construction (it saw >=1 finite e);
// the isfinite select is a 1-VALU defensive match of the reference's where().
// ---------------------------------------------------------------------------
__global__ void gat_edge_ex_kernel(float* __restrict__ e,
                                   const float* __restrict__ m,
                                   const int* __restrict__ dst,
                                   float* __restrict__ denom,
                                   int n_edges) {
    int i = blockIdx.x * blockDim.x + threadIdx.x;
    if (i >= n_edges) return;
    int d = dst[i];
    float mv = m[d];
    mv = isfinite(mv) ? mv : 0.0f;
    float ex = expf(e[i] - mv);
    e[i] = ex;
    unsafeAtomicAdd(&denom[d], ex);   // native global_atomic_add_f32
}

// ---------------------------------------------------------------------------
// Kernel 5: out[dst] += (ex/denom[dst]) * h[src]. One wave per edge:
// lane L owns columns 4L..4L+3 (one global_load_b128 + 4 f32 atomics).
// h/out are L2-resident (25.6 MB each vs 192 MB L2): L2-atomic-bound pass.
// ---------------------------------------------------------------------------
__global__ void gat_scatter_kernel(const float* __restrict__ h,
                                   const int* __restrict__ src,
                                   const int* __restrict__ dst,
                                   const float* __restrict__ ex,
                                   const float* __restrict__ denom,
                                   float* __restrict__ out,
                                   int n_edges) {
    int wave = (blockIdx.x * blockDim.x + threadIdx.x) >> 5;
    int lane = threadIdx.x & 31;
    if (wave >= n_edges) return;
    int s = src[wave];
    int d = dst[wave];
    float alpha = ex[wave] / denom[d];

    const float4* hp = (const float4*)(h + (size_t)s * D_DIM);
    float4 hv = hp[lane];                  // global_load_b128
    float* op = out + (size_t)d * D_DIM + lane * 4;
    unsafeAtomicAdd(op + 0, alpha * hv.x);
    unsafeAtomicAdd(op + 1, alpha * hv.y);
    unsafeAtomicAdd(op + 2, alpha * hv.z);
    unsafeAtomicAdd(op + 3, alpha * hv.w);
}

// ---------------------------------------------------------------------------
// Launch
// ---------------------------------------------------------------------------
extern "C" void kernel_launch(void* const* d_in, const int* in_sizes, int n_in,
                              void* d_out, int out_size, void* d_ws, size_t ws_size,
                              hipStream_t stream) {
    const float* h   = (const float*)d_in[0];
    const float* a_w = (const float*)d_in[1];
    const int*   src = (const int*)d_in[2];
    const int*   dst = (const int*)d_in[3];
    float*       out = (float*)d_out;

    const int n_nodes = in_sizes[0] / D_DIM;   // 50000
    const int n_edges = in_sizes[2];           // 800000

    // Workspace layout (floats): s1 | s2 | m | denom | e
    float* s1    = (float*)d_ws;
    float* s2    = s1 + n_nodes;
    float* m     = s2 + n_nodes;
    float* denom = m + n_nodes;
    float* e     = denom + n_nodes;

    const int BLK = 256;

    // 1) init
    {
        int total = out_size > n_nodes ? out_size : n_nodes;
        int grid = (total + BLK - 1) / BLK;
        gat_init_kernel<<<grid, BLK, 0, stream>>>(out, out_size, m, denom, n_nodes);
    }

    // 2) s1/s2 via WMMA: one wave per 16 nodes -> 8 waves (128 nodes) per block
    {
        int n_waves = (n_nodes + 15) / 16;
        int grid = (n_waves + (BLK / 32) - 1) / (BLK / 32);
        gat_gemv_wmma_kernel<<<grid, BLK, 0, stream>>>(h, a_w, s1, s2, n_nodes);
    }

    // 3) e + segment max
    {
        int grid = (n_edges + BLK - 1) / BLK;
        gat_edge_e_kernel<<<grid, BLK, 0, stream>>>(s1, s2, src, dst, e, m, n_edges);
    }

    // 4) ex + denom
    {
        int grid = (n_edges + BLK - 1) / BLK;
        gat_edge_ex_kernel<<<grid, BLK, 0, stream>>>(e, m, dst, denom, n_edges);
    }

    // 5) scatter: one wave per edge
    {
        int waves_per_blk = BLK / 32;
        int grid = (n_edges + waves_per_blk - 1) / waves_per_blk;
        gat_scatter_kernel<<<grid, BLK, 0, stream>>>(h, src, dst, e, denom, out, n_edges);
    }
}